// AnchorYOLO_18554258718788
// MI455X (gfx1250) — compile-verified
//
#include <hip/hip_runtime.h>
#include <hip/hip_bf16.h>
#include <stdint.h>

// ---------------- problem constants ----------------
#define NUM_C      80
#define TOPK_L     1000
#define NTOT       3000          // 3 * TOPK
#define NPAD       3072          // padded to multiple of 32 for IoU tiles
#define NSORT_G    4096          // global bitonic size
#define CAND_CAP   4096          // per-level candidate cap (32KB LDS sort)
#define HBINS      2048
#define CONF_T     0.05f
#define NMS_T      0.6f
#define IMG_SZ     2048.0f
#define CLS_OFF    1.0e5f
#define MASK_W     (NPAD / 32)   // 96 words per row

// ---------------- ws layout (bytes) ----------------
#define OFF_HIST   0u                         // 3*2048 u32
#define OFF_TBIN   24576u                     // 3 u32
#define OFF_CNT    24592u                     // 3 u32
#define OFF_CAND   24832u                     // 3*4096 u64
#define OFF_SCAT   123136u                    // 3000 f32 (per-level topk scores, conf-zeroed)
#define OFF_LCAT   135136u                    // 3000 u32 labels
#define OFF_BCAT   147136u                    // 3000*4 f32 boxes
#define OFF_OX1    195136u                    // 3072 f32 (offset boxes, sorted order)
#define OFF_OY1    207424u
#define OFF_OX2    219712u
#define OFF_OY2    232000u
#define OFF_AREA   244288u
#define OFF_SSORT  256576u                    // 3072 f32 sorted scores
#define OFF_MASK   268864u                    // 3072*96 u32 suppression mask

typedef __attribute__((ext_vector_type(2))) float v2f;
typedef __attribute__((ext_vector_type(4))) float v4f;
typedef __attribute__((ext_vector_type(8))) float v8f;

__device__ __forceinline__ float sigmoidf_(float x) {
    return 1.0f / (1.0f + __expf(-x));
}

__device__ __forceinline__ int score_bin(float s) {
    int b = (int)(s * (float)HBINS);
    return b < 0 ? 0 : (b > HBINS - 1 ? HBINS - 1 : b);
}

// ---------------- init scratch ----------------
__global__ void k_init(uint32_t* hist_cnt) {  // zeros hist + tbin + counters
    int n = 3 * HBINS + 8;
    for (int i = blockIdx.x * blockDim.x + threadIdx.x; i < n; i += gridDim.x * blockDim.x)
        hist_cnt[i] = 0u;
}

// ---------------- per-level score histogram (4 class-elems / thread) ----------------
// 80 % 4 == 0, so a 4-element group never straddles an anchor row: one obj
// sigmoid serves 4 scores, cls comes in as a single b128 load.
__global__ void k_hist(const float* __restrict__ obj, const float* __restrict__ cls,
                       int total4, uint32_t* __restrict__ hist) {
    __shared__ uint32_t h[HBINS];
    for (int i = threadIdx.x; i < HBINS; i += blockDim.x) h[i] = 0u;
    __syncthreads();
    for (int t = blockIdx.x * blockDim.x + threadIdx.x; t < total4; t += gridDim.x * blockDim.x) {
        int e0 = t * 4;
        int m  = e0 / NUM_C;
        float so = sigmoidf_(obj[m]);
        v4f c4 = *(const v4f*)(cls + e0);
#pragma unroll
        for (int q = 0; q < 4; ++q) {
            float s = sqrtf(so * sigmoidf_(c4[q]));
            atomicAdd(&h[score_bin(s)], 1u);
        }
    }
    __syncthreads();
    for (int i = threadIdx.x; i < HBINS; i += blockDim.x)
        if (h[i]) atomicAdd(&hist[i], h[i]);
}

// ---------------- pick threshold bin per level ----------------
__global__ void k_thresh(const uint32_t* __restrict__ hist, uint32_t* __restrict__ tbin) {
    int lvl = threadIdx.x;
    if (lvl >= 3) return;
    const uint32_t* h = hist + lvl * HBINS;
    uint32_t acc = 0; uint32_t e = 0;
    for (int b = HBINS - 1; b >= 0; --b) {
        acc += h[b];
        if (acc >= TOPK_L) { e = (uint32_t)b; break; }
    }
    tbin[lvl] = e;
}

// ---------------- collect candidates above threshold ----------------
__global__ void k_cand(const float* __restrict__ obj, const float* __restrict__ cls,
                       int total4, const uint32_t* __restrict__ tbin, int lvl,
                       unsigned long long* __restrict__ cand, uint32_t* __restrict__ cnt) {
    uint32_t tb = tbin[lvl];
    for (int t = blockIdx.x * blockDim.x + threadIdx.x; t < total4; t += gridDim.x * blockDim.x) {
        int e0 = t * 4;
        int m  = e0 / NUM_C;
        float so = sigmoidf_(obj[m]);
        v4f c4 = *(const v4f*)(cls + e0);
#pragma unroll
        for (int q = 0; q < 4; ++q) {
            float s = sqrtf(so * sigmoidf_(c4[q]));
            if ((uint32_t)score_bin(s) >= tb) {
                uint32_t pos = atomicAdd(cnt, 1u);
                if (pos < CAND_CAP)
                    cand[pos] = ((unsigned long long)__float_as_uint(s) << 32)
                              | (uint32_t)(~(uint32_t)(e0 + q));
            }
        }
    }
}

// ---------------- bitonic sort (descending) on shared u64 ----------------
__device__ __forceinline__ void bitonic_desc(unsigned long long* s, unsigned n,
                                             unsigned tid, unsigned nt) {
    for (unsigned k = 2; k <= n; k <<= 1) {
        for (unsigned j = k >> 1; j > 0; j >>= 1) {
            __syncthreads();
            for (unsigned i = tid; i < n; i += nt) {
                unsigned ixj = i ^ j;
                if (ixj > i) {
                    unsigned long long a = s[i], b = s[ixj];
                    bool descHalf = ((i & k) == 0);
                    if ((a < b) == descHalf) { s[i] = b; s[ixj] = a; }
                }
            }
        }
    }
    __syncthreads();
}

// ---------------- per-level top-k sort + box decode ----------------
__global__ void k_topk(const unsigned long long* __restrict__ cand,
                       const uint32_t* __restrict__ cnt,
                       const float* __restrict__ reg,
                       const float* __restrict__ anchor_size,
                       int lvl, int W, float stride,
                       float* __restrict__ scores_cat,
                       uint32_t* __restrict__ labels_cat,
                       float* __restrict__ boxes_cat) {
    __shared__ unsigned long long keys[CAND_CAP];
    unsigned tid = threadIdx.x, nt = blockDim.x;
    uint32_t c = *cnt; if (c > CAND_CAP) c = CAND_CAP;
    for (unsigned i = tid; i < CAND_CAP; i += nt)
        keys[i] = (i < c) ? cand[i] : 0ull;
    bitonic_desc(keys, CAND_CAP, tid, nt);

    if (tid < TOPK_L) {
        unsigned long long k = keys[tid];
        float sc = 0.0f; uint32_t idx = 0;
        if (k != 0ull) {
            sc  = __uint_as_float((uint32_t)(k >> 32));
            idx = ~(uint32_t)(k & 0xFFFFFFFFull);
        }
        uint32_t label  = idx % NUM_C;
        uint32_t anchor = idx / NUM_C;
        uint32_t cell = anchor / 3u, aIn = anchor % 3u;
        float ax = ((float)(cell % (uint32_t)W) + 0.5f) * stride;
        float ay = ((float)(cell / (uint32_t)W) + 0.5f) * stride;
        float aw = anchor_size[lvl * 6 + aIn * 2 + 0];
        float ah = anchor_size[lvl * 6 + aIn * 2 + 1];
        float r0 = reg[anchor * 4 + 0], r1 = reg[anchor * 4 + 1];
        float r2 = reg[anchor * 4 + 2], r3 = reg[anchor * 4 + 3];
        float cx = ax + (sigmoidf_(r0) * 3.0f - 1.5f) * stride;
        float cy = ay + (sigmoidf_(r1) * 3.0f - 1.5f) * stride;
        float bw = __expf(r2) * aw, bh = __expf(r3) * ah;
        int g = lvl * TOPK_L + (int)tid;
        scores_cat[g] = (sc > CONF_T) ? sc : 0.0f;   // conf filter (reference zeroes)
        labels_cat[g] = label;
        boxes_cat[g * 4 + 0] = cx - 0.5f * bw;
        boxes_cat[g * 4 + 1] = cy - 0.5f * bh;
        boxes_cat[g * 4 + 2] = cx + 0.5f * bw;
        boxes_cat[g * 4 + 3] = cy + 0.5f * bh;
    }
}

// ---------------- global sort + SoA prep + boxes/labels output ----------------
__global__ void k_gsort(const float* __restrict__ scores_cat,
                        const uint32_t* __restrict__ labels_cat,
                        const float* __restrict__ boxes_cat,
                        float* __restrict__ OX1, float* __restrict__ OY1,
                        float* __restrict__ OX2, float* __restrict__ OY2,
                        float* __restrict__ AREA, float* __restrict__ SS,
                        float* __restrict__ out) {
    __shared__ unsigned long long keys[NSORT_G];
    unsigned tid = threadIdx.x, nt = blockDim.x;
    for (unsigned i = tid; i < NSORT_G; i += nt) {
        if (i < NTOT) {
            float s = scores_cat[i];
            keys[i] = ((unsigned long long)__float_as_uint(s) << 32)
                    | (uint32_t)(~i);              // stable: ties -> lower pos first
        } else keys[i] = 0ull;
    }
    bitonic_desc(keys, NSORT_G, tid, nt);

    for (unsigned s = tid; s < NPAD; s += nt) {
        unsigned long long k = keys[s];
        float sc, b0, b1, b2, b3; int lab;
        if (k != 0ull) {
            uint32_t pos = ~(uint32_t)(k & 0xFFFFFFFFull);
            sc  = __uint_as_float((uint32_t)(k >> 32));
            lab = (int)labels_cat[pos];
            b0 = boxes_cat[pos * 4 + 0]; b1 = boxes_cat[pos * 4 + 1];
            b2 = boxes_cat[pos * 4 + 2]; b3 = boxes_cat[pos * 4 + 3];
        } else {                                    // synthetic pad: IoU-inert boxes
            sc = 0.0f; lab = 0;
            float base = 1.0e7f + 100.0f * (float)s;
            b0 = base; b1 = base; b2 = base + 1.0f; b3 = base + 1.0f;
        }
        if (s < NTOT) {
            out[s * 4 + 0] = fminf(fmaxf(b0 / IMG_SZ, 0.0f), 1.0f);
            out[s * 4 + 1] = fminf(fmaxf(b1 / IMG_SZ, 0.0f), 1.0f);
            out[s * 4 + 2] = fminf(fmaxf(b2 / IMG_SZ, 0.0f), 1.0f);
            out[s * 4 + 3] = fminf(fmaxf(b3 / IMG_SZ, 0.0f), 1.0f);
            out[4 * NTOT + NTOT + s] = (float)lab;  // labels after boxes+scores
        }
        float off = (float)lab * CLS_OFF;
        OX1[s] = b0 + off; OY1[s] = b1 + off;
        OX2[s] = b2 + off; OY2[s] = b3 + off;
        AREA[s] = (b2 - b0) * (b3 - b1);
        SS[s] = sc;
    }
}

// ---------------- IoU suppression mask: WMMA for all-pairs area sums ----------------
// One wave per 16(i) x 32(j) tile. V_WMMA_F32_16X16X4_F32 computes
// S[m][n] = area_i[m]*1 + 1*area_j[n] as rank-2 product; VALU does min/max
// intersection; ballot_w32 packs mask bits per C-layout row.
// iou > T is evaluated as inter > T*union (union > 0) -- no division chains.
__global__ void __launch_bounds__(256) k_ioumask(
        const float* __restrict__ OX1, const float* __restrict__ OY1,
        const float* __restrict__ OX2, const float* __restrict__ OY2,
        const float* __restrict__ AREA, uint32_t* __restrict__ mask) {
    const int wave = threadIdx.x >> 5;
    const int lane = threadIdx.x & 31;
    const int tile = blockIdx.x * 8 + wave;          // 18432 tiles total
    const int rowT = tile / MASK_W;                  // 0..191
    const int colT = tile % MASK_W;                  // 0..95
    const int i0 = rowT * 16, j0 = colT * 32;
    const int half = lane >> 4, n16 = lane & 15;

    // lanes 0..15 hold i-tile data at index = lane (16..31 load duplicates)
    float xi1l = OX1[i0 + n16], yi1l = OY1[i0 + n16];
    float xi2l = OX2[i0 + n16], yi2l = OY2[i0 + n16];
    float ail  = AREA[i0 + n16];

    // WMMA A operand (16x4 f32): K0 = area_i, K1 = 1, K2 = K3 = 0
    v2f a; a.x = half ? 0.0f : ail; a.y = half ? 0.0f : 1.0f;

    uint32_t bal[2][8];
#pragma unroll
    for (int sub = 0; sub < 2; ++sub) {
        int j = j0 + sub * 16 + n16;
        float xj1 = OX1[j], yj1 = OY1[j], xj2 = OX2[j], yj2 = OY2[j];
        float aj = AREA[j];
        // B operand (4x16 f32): row0 = 1, row1 = area_j, rows 2,3 = 0
        v2f b; b.x = half ? 0.0f : 1.0f; b.y = half ? 0.0f : aj;
        v8f c = {};
        c = __builtin_amdgcn_wmma_f32_16x16x4_f32(false, a, false, b,
                                                  (short)0, c, false, false);
#pragma unroll
        for (int r = 0; r < 8; ++r) {
            int iloc = r + 8 * half;
            float xi1 = __shfl(xi1l, iloc), yi1 = __shfl(yi1l, iloc);
            float xi2 = __shfl(xi2l, iloc), yi2 = __shfl(yi2l, iloc);
            float iw = fminf(xi2, xj2) - fmaxf(xi1, xj1);
            float ih = fminf(yi2, yj2) - fmaxf(yi1, yj1);
            float inter = fmaxf(iw, 0.0f) * fmaxf(ih, 0.0f);
            float uni = c[r] - inter + 1e-12f;       // always > 0
            bool pred = (inter > NMS_T * uni) && (j > i0 + iloc);
#if __has_builtin(__builtin_amdgcn_ballot_w32)
            bal[sub][r] = __builtin_amdgcn_ballot_w32(pred);
#else
            bal[sub][r] = (uint32_t)__ballot(pred);
#endif
        }
    }
    if (lane == 0) {
#pragma unroll
        for (int r = 0; r < 8; ++r) {
            uint32_t b0 = bal[0][r], b1 = bal[1][r];
            // C layout: vgpr r, lanes0-15 -> row r; lanes16-31 -> row r+8
            mask[(size_t)(i0 + r)     * MASK_W + colT] = (b0 & 0xFFFFu) | (b1 << 16);
            mask[(size_t)(i0 + r + 8) * MASK_W + colT] = (b0 >> 16) | (b1 & 0xFFFF0000u);
        }
    }
}

// ---------------- serial greedy NMS sweep (1 wave) ----------------
// Double-buffers the next mask row into LDS with global_load_async_to_lds_b32
// (ASYNCcnt-tracked, in-order), overlapping prefetch with the sweep.
__global__ void k_sweep(const uint32_t* __restrict__ mask,
                        const float* __restrict__ SS,
                        float* __restrict__ out_scores) {
    __shared__ uint32_t buf[2][MASK_W];
    const int lane = threadIdx.x;
    uint32_t s0 = 0, s1 = 0, s2 = 0;   // suppression bits: lane owns words l, 32+l, 64+l

    auto issue = [&](int row, int bsel) {
        uint32_t ldsbase = (uint32_t)(uintptr_t)(&buf[bsel][0]);
#pragma unroll
        for (int w = 0; w < 3; ++w) {
            uint32_t loff = ldsbase + (uint32_t)(lane + 32 * w) * 4u;
            unsigned long long ga =
                (unsigned long long)(uintptr_t)(mask + (size_t)row * MASK_W + lane + 32 * w);
            asm volatile("global_load_async_to_lds_b32 %0, %1, off"
                         :: "v"(loff), "v"(ga) : "memory");
        }
    };

    issue(0, 0);
    for (int i = 0; i < NTOT; ++i) {
        if (i + 1 < NTOT) {
            issue(i + 1, (i + 1) & 1);
            asm volatile("s_wait_asynccnt 0x3" ::: "memory");  // row i resident
        } else {
            asm volatile("s_wait_asynccnt 0x0" ::: "memory");
        }
        float si = SS[i];
        int wi = i >> 5;
        int slot = wi >> 5;                       // uniform
        uint32_t tmp = (slot == 0) ? s0 : (slot == 1) ? s1 : s2;
        uint32_t word = __shfl(tmp, wi & 31);
        bool suppressed = (word >> (i & 31)) & 1u;
        bool keep = (!suppressed) && (si > CONF_T);
        if (keep) {
            s0 |= buf[i & 1][lane];
            s1 |= buf[i & 1][lane + 32];
            s2 |= buf[i & 1][lane + 64];
        }
        if (lane == 0) out_scores[i] = keep ? si : 0.0f;
    }
}

// ---------------- launcher ----------------
extern "C" void kernel_launch(void* const* d_in, const int* in_sizes, int n_in,
                              void* d_out, int out_size, void* d_ws, size_t ws_size,
                              hipStream_t stream) {
    (void)in_sizes; (void)n_in; (void)out_size; (void)ws_size;
    const float* obj[3] = { (const float*)d_in[0], (const float*)d_in[3], (const float*)d_in[6] };
    const float* cls[3] = { (const float*)d_in[1], (const float*)d_in[4], (const float*)d_in[7] };
    const float* reg[3] = { (const float*)d_in[2], (const float*)d_in[5], (const float*)d_in[8] };
    const float* anc    = (const float*)d_in[9];
    const int   M[3]    = { 256 * 256 * 3, 128 * 128 * 3, 64 * 64 * 3 };
    const int   W[3]    = { 256, 128, 64 };
    const float strd[3] = { 8.0f, 16.0f, 32.0f };

    char* ws = (char*)d_ws;
    uint32_t* hist = (uint32_t*)(ws + OFF_HIST);
    uint32_t* tbin = (uint32_t*)(ws + OFF_TBIN);
    uint32_t* cnt  = (uint32_t*)(ws + OFF_CNT);
    unsigned long long* cand = (unsigned long long*)(ws + OFF_CAND);
    float*    scat = (float*)(ws + OFF_SCAT);
    uint32_t* lcat = (uint32_t*)(ws + OFF_LCAT);
    float*    bcat = (float*)(ws + OFF_BCAT);
    float* ox1 = (float*)(ws + OFF_OX1); float* oy1 = (float*)(ws + OFF_OY1);
    float* ox2 = (float*)(ws + OFF_OX2); float* oy2 = (float*)(ws + OFF_OY2);
    float* area = (float*)(ws + OFF_AREA);
    float* ssort = (float*)(ws + OFF_SSORT);
    uint32_t* msk = (uint32_t*)(ws + OFF_MASK);
    float* out = (float*)d_out;

    k_init<<<32, 256, 0, stream>>>(hist);
    for (int l = 0; l < 3; ++l)
        k_hist<<<2048, 256, 0, stream>>>(obj[l], cls[l], M[l] * NUM_C / 4, hist + l * HBINS);
    k_thresh<<<1, 64, 0, stream>>>(hist, tbin);
    for (int l = 0; l < 3; ++l)
        k_cand<<<2048, 256, 0, stream>>>(obj[l], cls[l], M[l] * NUM_C / 4, tbin, l,
                                         cand + l * CAND_CAP, cnt + l);
    for (int l = 0; l < 3; ++l)
        k_topk<<<1, 1024, 0, stream>>>(cand + l * CAND_CAP, cnt + l, reg[l], anc,
                                       l, W[l], strd[l], scat, lcat, bcat);
    k_gsort<<<1, 1024, 0, stream>>>(scat, lcat, bcat, ox1, oy1, ox2, oy2, area, ssort, out);
    k_ioumask<<<(192 * MASK_W) / 8, 256, 0, stream>>>(ox1, oy1, ox2, oy2, area, msk);
    k_sweep<<<1, 32, 0, stream>>>(msk, ssort, out + 4 * NTOT);
}